// TensorSnake_34239479283737
// MI455X (gfx1250) — compile-verified
//
#include <hip/hip_runtime.h>

#define BOARD 64
#define CELLS 4096          // 64*64
#define TPB   256           // 8 wave32 per block
#define V4PT  4             // float4 loads per thread (4*4*256 = 4096 cells)

#if defined(__has_builtin)
#  if __has_builtin(__builtin_amdgcn_global_load_async_to_lds_b128) && \
      __has_builtin(__builtin_amdgcn_s_wait_asynccnt)
#    define USE_ASYNC_LDS 1
#  endif
#endif
#ifndef USE_ASYNC_LDS
#  define USE_ASYNC_LDS 0
#endif

#if USE_ASYNC_LDS
// Parameter type revealed by the round-1 diagnostic:
//   '__attribute__((__vector_size__(4 * sizeof(int)))) int __device__ *'
typedef __attribute__((vector_size(16))) int vi4;
typedef __attribute__((address_space(1))) vi4 gvi4;   // global b128
typedef __attribute__((address_space(3))) vi4 lvi4;   // LDS b128
#endif

__global__ __launch_bounds__(TPB) void snake_step_kernel(
    const float* __restrict__ state,
    const int*   __restrict__ pos_prev,
    const int*   __restrict__ pos_cur,
    const int*   __restrict__ action,
    const float* __restrict__ noise,
    float*       __restrict__ out)
{
    const int g   = blockIdx.x;
    const int tid = threadIdx.x;
    const long gbase = (long)g * CELLS;
    const float* __restrict__ st = state + gbase;
    const float* __restrict__ nz = noise + gbase;
    float*       __restrict__ op = out   + gbase;

#if USE_ASYNC_LDS
    __shared__ float lnoise[CELLS];
    {
        // Kick off async noise -> LDS copies first; they overlap with the
        // state loads and per-game scalar logic below (ASYNCcnt).
        #pragma unroll
        for (int k = 0; k < V4PT; ++k) {
            int v4 = k * TPB + tid;
            __builtin_amdgcn_global_load_async_to_lds_b128(
                (gvi4*)(nz + v4 * 4),
                (lvi4*)(lnoise + v4 * 4),
                0, 0);
        }
    }
#endif

    // ---- vector loads of this thread's 16 state cells (coalesced b128) ----
    float4 sv[V4PT];
    #pragma unroll
    for (int k = 0; k < V4PT; ++k) {
        int v4 = k * TPB + tid;
        sv[k] = ((const float4*)st)[v4];
    }

    // ---- per-game scalar logic (uniform per block; compiler scalarizes) ----
    int ppx = pos_prev[g * 2 + 0], ppy = pos_prev[g * 2 + 1];
    int pcx = pos_cur [g * 2 + 0], pcy = pos_cur [g * 2 + 1];
    int a   = action[g];
    int dx = pcx - ppx, dy = pcy - ppy;
    int ndx = dx, ndy = dy;
    if (a == 0)      { ndx = -dy; ndy =  dx; }   // left
    else if (a == 2) { ndx =  dy; ndy = -dx; }   // right
    int nx = pcx + ndx, ny = pcy + ndy;
    bool outside = (nx < 0) | (nx >= BOARD) | (ny < 0) | (ny >= BOARD);
    nx = min(max(nx, 0), BOARD - 1);
    ny = min(max(ny, 0), BOARD - 1);
    float cell   = st[nx * BOARD + ny];
    bool dead    = outside | (cell > 0.0f);
    bool feeding = (cell == -1.0f);
    const int c = BOARD / 2;
    if (dead) { pcx = c; pcy = c - 1; nx = c; ny = c; }
    bool spawn = dead | feeding;
    const int pn_idx = nx * BOARD + ny;
    const int pc_idx = pcx * BOARD + pcy;
    const int r0 = c * BOARD + (c - 2);   // 2078 -> 1.0
    const int r1 = r0 + 1;                // 2079 -> 2.0
    const int r2 = r0 + 2;                // 2080 -> 1.0

    // ---- pass 1: intermediate board (zero-on-dead + reset cells),
    //      masked-argmax candidates from noise ----
    float v[V4PT * 4];
    float best    = -__builtin_inff();
    int   bestIdx = 0x7FFFFFFF;

#if USE_ASYNC_LDS
    __builtin_amdgcn_s_wait_asynccnt(0);
    __syncthreads();
#endif

    #pragma unroll
    for (int k = 0; k < V4PT; ++k) {
        int v4 = k * TPB + tid;
#if USE_ASYNC_LDS
        float4 nv = ((const float4*)lnoise)[v4];
#else
        float4 nv = ((const float4*)nz)[v4];
#endif
        #pragma unroll
        for (int e = 0; e < 4; ++e) {
            int   idx = v4 * 4 + e;
            float s   = (&sv[k].x)[e];
            float val = dead ? 0.0f : s;
            if (dead) {
                if      (idx == r0) val = 1.0f;
                else if (idx == r1) val = 2.0f;
                else if (idx == r2) val = 1.0f;
            }
            v[k * 4 + e] = val;
            float cand = (val == 0.0f) ? (&nv.x)[e] : -__builtin_inff();
            if (cand > best || (cand == best && idx < bestIdx)) {
                best = cand; bestIdx = idx;
            }
        }
    }

    // ---- block argmax reduction (wave32 shuffle, then LDS across 8 waves),
    //      first-index tie-break to match jnp.argmax ----
    #pragma unroll
    for (int off = 16; off > 0; off >>= 1) {
        float ov = __shfl_xor(best, off, 32);
        int   oi = __shfl_xor(bestIdx, off, 32);
        if (ov > best || (ov == best && oi < bestIdx)) { best = ov; bestIdx = oi; }
    }
    __shared__ float swv[TPB / 32];
    __shared__ int   swi[TPB / 32];
    __shared__ int   sfood;
    const int wave = tid >> 5, lane = tid & 31;
    if (lane == 0) { swv[wave] = best; swi[wave] = bestIdx; }
    __syncthreads();
    if (tid == 0) {
        float b = swv[0]; int bi = swi[0];
        #pragma unroll
        for (int w = 1; w < TPB / 32; ++w) {
            if (swv[w] > b || (swv[w] == b && swi[w] < bi)) { b = swv[w]; bi = swi[w]; }
        }
        sfood = bi;
    }
    __syncthreads();
    const int food = sfood;

    // ---- head = value at pos_cur after food placement + decrement ----
    float pcv = dead ? 2.0f : st[pc_idx];
    if (spawn && pc_idx == food) pcv = -1.0f;
    if (pcv > 0.0f && !feeding)  pcv -= 1.0f;
    const float head = pcv;

    // ---- pass 2: food spawn, decrement, head write; b128 stores ----
    #pragma unroll
    for (int k = 0; k < V4PT; ++k) {
        int v4 = k * TPB + tid;
        float4 o;
        #pragma unroll
        for (int e = 0; e < 4; ++e) {
            int   idx = v4 * 4 + e;
            float val = v[k * 4 + e];
            if (spawn && idx == food)   val = -1.0f;
            if (val > 0.0f && !feeding) val -= 1.0f;
            if (idx == pn_idx)          val = head + 1.0f;
            (&o.x)[e] = val;
        }
        ((float4*)op)[v4] = o;
    }
}

extern "C" void kernel_launch(void* const* d_in, const int* in_sizes, int n_in,
                              void* d_out, int out_size, void* d_ws, size_t ws_size,
                              hipStream_t stream) {
    const float* state    = (const float*)d_in[0];
    const int*   pos_prev = (const int*)  d_in[1];
    const int*   pos_cur  = (const int*)  d_in[2];
    const int*   action   = (const int*)  d_in[3];
    const float* noise    = (const float*)d_in[4];
    float*       out      = (float*)d_out;
    const int games = in_sizes[3];   // action has one entry per game
    snake_step_kernel<<<games, TPB, 0, stream>>>(state, pos_prev, pos_cur,
                                                 action, noise, out);
}